// GCNModel_55817394978940
// MI455X (gfx1250) — compile-verified
//
#include <hip/hip_runtime.h>
#include <hip/hip_bf16.h>
#include <cstdint>

typedef __attribute__((ext_vector_type(2))) float v2f;
typedef __attribute__((ext_vector_type(4))) float v4f;
typedef __attribute__((ext_vector_type(8))) float v8f;

// Compile-time feature sizes from the reference model
#define IN_F_C  256
#define GNN_F_C 128
#define OUT_F_C 64

// ---------------------------------------------------------------------------
// out[N, NOUT] = x[N, K] @ W[NOUT, K]^T + b      (PyTorch Linear layout)
// One wave computes one 16x16 output tile via V_WMMA_F32_16X16X4_F32.
// Block = (NOUT/16) waves covering the full output width of one 16-row slab.
//
// f32 WMMA operand layout (CDNA5 ISA):
//   A (16x4):  lane L -> row m = L%16 ; VGPR0 = A[m][2*(L/16)], VGPR1 = A[m][2*(L/16)+1]
//   B (4x16):  lane L -> col n = L%16 ; VGPR0 = B[2*(L/16)][n], VGPR1 = B[2*(L/16)+1][n]
//   C (16x16): lane L -> col n = L%16 ; VGPR r = C[r + 8*(L/16)][n]
// B[k][n] = W[n][k] so both A and B are contiguous float2 loads.
// ---------------------------------------------------------------------------
template<int K, int NOUT>
__global__ __launch_bounds__((NOUT / 16) * 32)
void linear_wmma(const float* __restrict__ x, const float* __restrict__ W,
                 const float* __restrict__ bias, float* __restrict__ out,
                 int nrows)
{
    const int lane = threadIdx.x & 31;
    const int wave = threadIdx.x >> 5;       // which 16-col tile
    const int m    = lane & 15;
    const int half = lane >> 4;              // 0 or 1 (k-half / row-half)
    const int rowBase = blockIdx.x * 16;
    const int n = wave * 16 + m;             // output column for this lane

    int ar = rowBase + m;                    // A row (clamp tail, store-guarded)
    if (ar >= nrows) ar = nrows - 1;

    const float* __restrict__ aptr = x + (size_t)ar * K + 2 * half;
    const float* __restrict__ bptr = W + (size_t)n  * K + 2 * half;

    v8f acc = {};
    #pragma unroll 8
    for (int k = 0; k < K; k += 4) {
        v2f a = *(const v2f*)(aptr + k);
        v2f b = *(const v2f*)(bptr + k);
        // 8 args: (neg_a, A, neg_b, B, c_mod, C, reuse_a, reuse_b)
        acc = __builtin_amdgcn_wmma_f32_16x16x4_f32(
            false, a, false, b, (short)0, acc, false, false);
    }

    const float bv = bias[n];
    float* __restrict__ optr = out + (size_t)(rowBase + half * 8) * NOUT + n;
    if (rowBase + 16 <= nrows) {
        // fast path: whole tile in range, straight-line stores (no exec juggling)
        #pragma unroll
        for (int r = 0; r < 8; ++r)
            optr[(size_t)r * NOUT] = acc[r] + bv;
    } else {
        #pragma unroll
        for (int r = 0; r < 8; ++r)
            if (rowBase + half * 8 + r < nrows)
                optr[(size_t)r * NOUT] = acc[r] + bv;
    }
}

// ---------------------------------------------------------------------------
// agg[dst[e], f..f+3] += h[src[e], f..f+3]
// Grid-stride over E*(F/4): one b128 gather + 4 global_atomic_add_f32 per
// thread. agg fits in the 192 MB L2 so atomics resolve in L2.
// ---------------------------------------------------------------------------
template<int F>
__global__ void scatter_add(const float* __restrict__ h,
                            const int* __restrict__ src,
                            const int* __restrict__ dst,
                            float* __restrict__ agg, long long total4)
{
    constexpr int Q = F / 4;                 // float4 chunks per row
    long long i = (long long)blockIdx.x * blockDim.x + threadIdx.x;
    const long long stride = (long long)gridDim.x * blockDim.x;
    for (; i < total4; i += stride) {
        const long long e = i / Q;           // Q is a power of two -> shift
        const int q = (int)(i - e * Q);
        const int s = src[e];
        const int d = dst[e];
        const v4f v = *(const v4f*)(h + (size_t)s * F + 4 * q);
        float* p = agg + (size_t)d * F + 4 * q;
        atomicAdd(p + 0, v.x);
        atomicAdd(p + 1, v.y);
        atomicAdd(p + 2, v.z);
        atomicAdd(p + 3, v.w);
    }
}

// h[i, f..f+3] = (relu?) max(0, norm[i] * h[i, f..f+3]) : in place, float4
template<int F, bool RELU>
__global__ void norm_mul(float* __restrict__ h, const float* __restrict__ norm,
                         long long total4)
{
    constexpr int Q = F / 4;
    long long i = (long long)blockIdx.x * blockDim.x + threadIdx.x;
    const long long stride = (long long)gridDim.x * blockDim.x;
    for (; i < total4; i += stride) {
        const long long row = i / Q;
        const float s = norm[row];
        v4f v = *(const v4f*)(h + 4 * i);
        v.x *= s; v.y *= s; v.z *= s; v.w *= s;
        if (RELU) {
            v.x = v.x > 0.0f ? v.x : 0.0f;
            v.y = v.y > 0.0f ? v.y : 0.0f;
            v.z = v.z > 0.0f ? v.z : 0.0f;
            v.w = v.w > 0.0f ? v.w : 0.0f;
        }
        *(v4f*)(h + 4 * i) = v;
    }
}

extern "C" void kernel_launch(void* const* d_in, const int* in_sizes, int n_in,
                              void* d_out, int out_size, void* d_ws, size_t ws_size,
                              hipStream_t stream)
{
    const float* x    = (const float*)d_in[0];
    const float* norm = (const float*)d_in[1];
    const int*   src  = (const int*)d_in[2];
    const int*   dst  = (const int*)d_in[3];
    const float* W1   = (const float*)d_in[4];
    const float* b1   = (const float*)d_in[5];
    const float* W2   = (const float*)d_in[6];
    const float* b2   = (const float*)d_in[7];

    const int N = in_sizes[1];   // norm has N elements
    const int E = in_sizes[2];

    float* h1   = (float*)d_ws;                    // N * GNN_F
    float* agg1 = h1 + (size_t)N * GNN_F_C;        // N * GNN_F (then relu/norm in place)
    float* h2   = h1;                              // reuse: h1 dead after scatter 1
    float* out  = (float*)d_out;                   // N * OUT_F

    const int rowTiles = (N + 15) / 16;
    const int sblocks  = 16384;                    // grid-stride scatter blocks
    const long long e4_1 = (long long)E * (GNN_F_C / 4);
    const long long e4_2 = (long long)E * (OUT_F_C / 4);
    const long long n4_1 = (long long)N * (GNN_F_C / 4);
    const long long n4_2 = (long long)N * (OUT_F_C / 4);

    // ---- Layer 1: h1 = x @ W1^T + b1 ----
    hipMemsetAsync(agg1, 0, (size_t)N * GNN_F_C * sizeof(float), stream);
    linear_wmma<IN_F_C, GNN_F_C>
        <<<rowTiles, (GNN_F_C / 16) * 32, 0, stream>>>(x, W1, b1, h1, N);
    scatter_add<GNN_F_C>
        <<<sblocks, 256, 0, stream>>>(h1, src, dst, agg1, e4_1);
    norm_mul<GNN_F_C, true>
        <<<(int)((n4_1 + 255) / 256), 256, 0, stream>>>(agg1, norm, n4_1);

    // ---- Layer 2: h2 = relu(norm*agg1) @ W2^T + b2 ----
    linear_wmma<GNN_F_C, OUT_F_C>
        <<<rowTiles, (OUT_F_C / 16) * 32, 0, stream>>>(agg1, W2, b2, h2, N);
    hipMemsetAsync(out, 0, (size_t)N * OUT_F_C * sizeof(float), stream);
    scatter_add<OUT_F_C>
        <<<sblocks, 256, 0, stream>>>(h2, src, dst, out, e4_2);
    norm_mul<OUT_F_C, false>
        <<<(int)((n4_2 + 255) / 256), 256, 0, stream>>>(out, norm, n4_2);
}